// MyLightGCNWithAttn_38817914421716
// MI455X (gfx1250) — compile-verified
//
#include <hip/hip_runtime.h>

typedef __attribute__((ext_vector_type(2))) float v2f;
typedef __attribute__((ext_vector_type(4))) float v4f;
typedef __attribute__((ext_vector_type(8))) float v8f;

#define N_USER    50000
#define N_ITEM    10000
#define EMBED_DIM 64
#define NUM_NODES (N_USER + N_ITEM)
#define ALPHA0    (1.0f / 3.0f)

// ---------------------------------------------------------------------------
// Kernel 1: build fused node table  out[node][d]  (ALPHA0 folded in).
// One float4 per thread, 16 threads per node -> contiguous 256B per node.
// ---------------------------------------------------------------------------
__global__ void fuse_nodes_kernel(const float* __restrict__ user_emb,
                                  const float* __restrict__ item_emb,
                                  const float* __restrict__ tag_emb,
                                  const float* __restrict__ testid_emb,
                                  const float* __restrict__ bigcat_emb,
                                  const float* __restrict__ daydiff_emb,
                                  const int*   __restrict__ item_tags,
                                  const int*   __restrict__ item_testids,
                                  const int*   __restrict__ item_bigcat,
                                  const int*   __restrict__ user_daydiff,
                                  float*       __restrict__ table)
{
    int gid  = blockIdx.x * blockDim.x + threadIdx.x;
    int node = gid >> 4;
    int c4   = (gid & 15) << 2;           // float offset within the 64-dim row
    if (node >= NUM_NODES) return;

    v4f r;
    if (node < N_USER) {
        int dd = user_daydiff[node];
        v4f u = *(const v4f*)(user_emb    + (size_t)node * EMBED_DIM + c4);
        v4f w = *(const v4f*)(daydiff_emb + (size_t)dd   * EMBED_DIM + c4);
        r = (u + w) * (0.5f * ALPHA0);
    } else {
        int j = node - N_USER;
        v4f a = *(const v4f*)(item_emb   + (size_t)j                  * EMBED_DIM + c4);
        v4f b = *(const v4f*)(tag_emb    + (size_t)item_tags[j]       * EMBED_DIM + c4);
        v4f c = *(const v4f*)(testid_emb + (size_t)item_testids[j]    * EMBED_DIM + c4);
        v4f d = *(const v4f*)(bigcat_emb + (size_t)item_bigcat[j]     * EMBED_DIM + c4);
        r = (a + b + c + d) * (0.25f * ALPHA0);
    }
    *(v4f*)(table + (size_t)node * EMBED_DIM + c4) = r;
}

// ---------------------------------------------------------------------------
// Helper: load one 32-float half of the fused node vector into 8 float4 regs.
// USE_TABLE=true reads the precomputed table; false recomputes inline.
// ---------------------------------------------------------------------------
template <bool USE_TABLE>
__device__ __forceinline__ void load_half(int node, int half, v4f* __restrict__ r,
                                          const float* __restrict__ table,
                                          const float* __restrict__ user_emb,
                                          const float* __restrict__ item_emb,
                                          const float* __restrict__ tag_emb,
                                          const float* __restrict__ testid_emb,
                                          const float* __restrict__ bigcat_emb,
                                          const float* __restrict__ daydiff_emb,
                                          const int*   __restrict__ item_tags,
                                          const int*   __restrict__ item_testids,
                                          const int*   __restrict__ item_bigcat,
                                          const int*   __restrict__ user_daydiff)
{
    int c0 = half * 32;
    if (USE_TABLE) {
        const v4f* p = (const v4f*)(table + (size_t)node * EMBED_DIM + c0);
#pragma unroll
        for (int i = 0; i < 8; ++i) r[i] = p[i];
    } else {
        if (node < N_USER) {
            int dd = user_daydiff[node];
            const v4f* pu = (const v4f*)(user_emb    + (size_t)node * EMBED_DIM + c0);
            const v4f* pw = (const v4f*)(daydiff_emb + (size_t)dd   * EMBED_DIM + c0);
#pragma unroll
            for (int i = 0; i < 8; ++i) r[i] = (pu[i] + pw[i]) * (0.5f * ALPHA0);
        } else {
            int j = node - N_USER;
            const v4f* pa = (const v4f*)(item_emb   + (size_t)j               * EMBED_DIM + c0);
            const v4f* pb = (const v4f*)(tag_emb    + (size_t)item_tags[j]    * EMBED_DIM + c0);
            const v4f* pc = (const v4f*)(testid_emb + (size_t)item_testids[j] * EMBED_DIM + c0);
            const v4f* pd = (const v4f*)(bigcat_emb + (size_t)item_bigcat[j]  * EMBED_DIM + c0);
#pragma unroll
            for (int i = 0; i < 8; ++i) r[i] = (pa[i] + pb[i] + pc[i] + pd[i]) * (0.25f * ALPHA0);
        }
    }
}

// ---------------------------------------------------------------------------
// Kernel 2: per-edge dot products via V_WMMA_F32_16X16X4_F32.
// One wave (32 lanes) handles 16 edges. Lane L owns edge (L&15), half (L>>4)
// of its 64-float vector. WMMA step s feeds logical K{0,1} from elements
// {2s,2s+1} (lanes 0-15) and K{2,3} from {32+2s,32+2s+1} (lanes 16-31),
// identically for A (src) and B (dst) -> C = Src x Dst^T exactly (fp32).
// Diagonal C[e][e] = dot(src_e, dst_e).
// ---------------------------------------------------------------------------
template <bool USE_TABLE>
__global__ void edge_dot_wmma_kernel(const int* __restrict__ e_src,
                                     const int* __restrict__ e_dst,
                                     const float* __restrict__ table,
                                     const float* __restrict__ user_emb,
                                     const float* __restrict__ item_emb,
                                     const float* __restrict__ tag_emb,
                                     const float* __restrict__ testid_emb,
                                     const float* __restrict__ bigcat_emb,
                                     const float* __restrict__ daydiff_emb,
                                     const int*   __restrict__ item_tags,
                                     const int*   __restrict__ item_testids,
                                     const int*   __restrict__ item_bigcat,
                                     const int*   __restrict__ user_daydiff,
                                     float* __restrict__ out,
                                     int nEdges, int nTiles)
{
    int wave = (blockIdx.x * blockDim.x + threadIdx.x) >> 5;
    int lane = threadIdx.x & 31;
    if (wave >= nTiles) return;          // wave-uniform: EXEC stays all-1s below

    int m    = lane & 15;                // edge within tile
    int half = lane >> 4;                // which 32-float half this lane owns
    int e    = wave * 16 + m;
    int ec   = min(e, nEdges - 1);       // clamp tail loads; store predicated later

    int sIdx = e_src[ec];
    int dIdx = e_dst[ec];

    v4f ra[8], rb[8];
    load_half<USE_TABLE>(sIdx, half, ra, table, user_emb, item_emb, tag_emb,
                         testid_emb, bigcat_emb, daydiff_emb,
                         item_tags, item_testids, item_bigcat, user_daydiff);
    load_half<USE_TABLE>(dIdx, half, rb, table, user_emb, item_emb, tag_emb,
                         testid_emb, bigcat_emb, daydiff_emb,
                         item_tags, item_testids, item_bigcat, user_daydiff);

    v8f acc = {};
#pragma unroll
    for (int s = 0; s < 16; ++s) {
        v4f va = ra[s >> 1], vb = rb[s >> 1];
        v2f a, b;
        if (s & 1) { a[0] = va[2]; a[1] = va[3]; b[0] = vb[2]; b[1] = vb[3]; }
        else       { a[0] = va[0]; a[1] = va[1]; b[0] = vb[0]; b[1] = vb[1]; }
        acc = __builtin_amdgcn_wmma_f32_16x16x4_f32(
            /*neg_a=*/false, a, /*neg_b=*/false, b,
            /*c_mod=*/(short)0, acc, /*reuse_a=*/false, /*reuse_b=*/false);
    }

    // Diagonal extraction: element (m,m) lives in VGPR (m&7) on lane m (m<8)
    // or lane m+16 (m>=8)  =>  producer lanes are 0-7 and 24-31, component lane&7.
    int t = lane & 7;
    float d01 = (t & 1) ? acc[1] : acc[0];
    float d23 = (t & 1) ? acc[3] : acc[2];
    float d45 = (t & 1) ? acc[5] : acc[4];
    float d67 = (t & 1) ? acc[7] : acc[6];
    float d03 = (t & 2) ? d23 : d01;
    float d47 = (t & 2) ? d67 : d45;
    float diag = (t & 4) ? d47 : d03;

    bool producer = (lane < 8) || (lane >= 24);
    if (producer && e < nEdges) out[e] = diag;
}

// ---------------------------------------------------------------------------
extern "C" void kernel_launch(void* const* d_in, const int* in_sizes, int n_in,
                              void* d_out, int out_size, void* d_ws, size_t ws_size,
                              hipStream_t stream)
{
    const float* user_emb    = (const float*)d_in[0];
    const float* item_emb    = (const float*)d_in[1];
    const float* tag_emb     = (const float*)d_in[2];
    const float* testid_emb  = (const float*)d_in[3];
    const float* bigcat_emb  = (const float*)d_in[4];
    const float* daydiff_emb = (const float*)d_in[5];
    const int*   edge_index  = (const int*)d_in[6];
    const int*   item_tags    = (const int*)d_in[7];
    const int*   item_testids = (const int*)d_in[8];
    const int*   item_bigcat  = (const int*)d_in[9];
    const int*   user_daydiff = (const int*)d_in[10];
    float* out = (float*)d_out;

    int nEdges = in_sizes[6] / 2;
    const int* e_src = edge_index;
    const int* e_dst = edge_index + nEdges;

    int nTiles = (nEdges + 15) / 16;
    int wavesPerBlock = 8;                       // 256 threads = 8 wave32
    int edgeBlocks = (nTiles + wavesPerBlock - 1) / wavesPerBlock;

    size_t tableBytes = (size_t)NUM_NODES * EMBED_DIM * sizeof(float);

    if (ws_size >= tableBytes) {
        float* table = (float*)d_ws;
        int nThreads = NUM_NODES * 16;
        fuse_nodes_kernel<<<(nThreads + 255) / 256, 256, 0, stream>>>(
            user_emb, item_emb, tag_emb, testid_emb, bigcat_emb, daydiff_emb,
            item_tags, item_testids, item_bigcat, user_daydiff, table);
        edge_dot_wmma_kernel<true><<<edgeBlocks, 256, 0, stream>>>(
            e_src, e_dst, table, user_emb, item_emb, tag_emb, testid_emb,
            bigcat_emb, daydiff_emb, item_tags, item_testids, item_bigcat,
            user_daydiff, out, nEdges, nTiles);
    } else {
        // Workspace too small: recompute fused embeddings inline per edge.
        edge_dot_wmma_kernel<false><<<edgeBlocks, 256, 0, stream>>>(
            e_src, e_dst, nullptr, user_emb, item_emb, tag_emb, testid_emb,
            bigcat_emb, daydiff_emb, item_tags, item_testids, item_bigcat,
            user_daydiff, out, nEdges, nTiles);
    }
}